// GNN2_46437186404821
// MI455X (gfx1250) — compile-verified
//
#include <hip/hip_runtime.h>
#include <math.h>

#define DFEAT 64

typedef __attribute__((ext_vector_type(2))) float v2f;
typedef __attribute__((ext_vector_type(8))) float v8f;

// Native fp32 L2 atomic add, device scope (guaranteed: no CAS-loop fallback).
__device__ __forceinline__ void atomic_add_f32(float* p, float v) {
    asm volatile("global_atomic_add_f32 %0, %1, off scope:SCOPE_DEV"
                 : : "v"(p), "v"(v) : "memory");
}

// ---------------- utility ----------------
__global__ void zero_f32x4(float4* __restrict__ p, int n4) {
    int i = blockIdx.x * blockDim.x + threadIdx.x;
    if (i < n4) p[i] = make_float4(0.f, 0.f, 0.f, 0.f);
}

// ---------------- edge phase ----------------
// denom[dst] += adv  (scalar fp32 atomics, resolve in L2)
__global__ void edge_denom(const int* __restrict__ dst, const float* __restrict__ adv,
                           float* __restrict__ denom, int E) {
    int e = blockIdx.x * blockDim.x + threadIdx.x;
    if (e < E) atomic_add_f32(&denom[dst[e]], adv[e]);
}

// aggr[dst] += x[src] * (adv/denom[dst]) — 16 threads/edge.
// Leader lane of each 16-lane group loads indices + computes att once, then
// broadcasts via shfl. float4 gather + 4 native fp32 atomics per thread.
// x and aggr are L2-resident (25.6 MB each << 192 MB L2).
__global__ void edge_scatter(const int* __restrict__ src, const int* __restrict__ dst,
                             const float* __restrict__ adv, const float* __restrict__ denom,
                             const float* __restrict__ x, float* __restrict__ aggr, int E) {
    int idx = blockIdx.x * blockDim.x + threadIdx.x;
    int e = idx >> 4;
    if (e >= E) return;
    int lane = threadIdx.x & 31;
    int leader = lane & 16;                 // lane 0 or 16 leads its 16-lane group
    int s = 0, d = 0; float a = 0.f;
    if ((lane & 15) == 0) {
        s = src[e];
        d = dst[e];
        a = adv[e] / denom[d];
    }
    s = __shfl(s, leader, 32);
    d = __shfl(d, leader, 32);
    a = __shfl(a, leader, 32);

    int q = (idx & 15) << 2;
    const float4 xv = *(const float4*)(x + (size_t)s * DFEAT + q);
    float* ap = aggr + (size_t)d * DFEAT + q;
    atomic_add_f32(ap + 0, xv.x * a);
    atomic_add_f32(ap + 1, xv.y * a);
    atomic_add_f32(ap + 2, xv.z * a);
    atomic_add_f32(ap + 3, xv.w * a);
}

// ---------------- node phase (fully fused) ----------------
// out = LayerNorm( (gelu(aggr)+x) @ W + b ) * g + be
// One wave32 per 16-row tile; fp32 WMMA 16x16x4. gelu+residual fused into the
// A-fragment load (each h element produced exactly once). W staged in LDS in
// K-pair-interleaved v2f layout so each B fragment is one ds_load_b64.
__global__ void __launch_bounds__(256)
gemm_bias_ln(const float* __restrict__ aggr, const float* __restrict__ xin,
             const float* __restrict__ W, const float* __restrict__ bias,
             const float* __restrict__ g, const float* __restrict__ be,
             float* __restrict__ out, int N) {
    __shared__ v2f   Wp[32 * 64];                // Wp[k2*64+n] = {W[2k2][n], W[2k2+1][n]}
    __shared__ float tile[8][16][DFEAT + 4];     // per-wave result tile, padded

    int tid = threadIdx.x;
    for (int i = tid; i < 32 * 64; i += 256) {
        int k2 = i >> 6, n = i & 63;
        v2f w;
        w.x = W[(2 * k2) * DFEAT + n];
        w.y = W[(2 * k2 + 1) * DFEAT + n];
        Wp[i] = w;
    }
    __syncthreads();

    int wave = tid >> 5;
    int lane = tid & 31;
    long rowBase = ((long)blockIdx.x * 8 + wave) * 16;
    if (rowBase >= N) return;                    // wave-uniform: EXEC stays all-1s

    int m  = lane & 15;                          // M index (A) / N index (B,C)
    int kh = (lane >> 4) << 1;                   // 0 for lanes 0-15, 2 for lanes 16-31

    long arow = rowBase + m; if (arow >= N) arow = N - 1;   // clamp keeps EXEC full
    const float* ag = aggr + arow * DFEAT;
    const float* xr = xin  + arow * DFEAT;

    v8f c0 = {}, c1 = {}, c2 = {}, c3 = {};
    #pragma unroll
    for (int k0 = 0; k0 < DFEAT; k0 += 4) {
        int k = k0 + kh;
        // fused: h = gelu(aggr) + x  (exact gelu via erf)
        float v0 = ag[k], v1 = ag[k + 1];
        v2f a;
        a.x = 0.5f * v0 * (1.0f + erff(v0 * 0.70710678118654752f)) + xr[k];
        a.y = 0.5f * v1 * (1.0f + erff(v1 * 0.70710678118654752f)) + xr[k + 1];

        const v2f* wrow = Wp + ((k0 >> 1) + (lane >> 4)) * 64;
        v2f b0 = wrow[m];
        v2f b1 = wrow[16 + m];
        v2f b2 = wrow[32 + m];
        v2f b3 = wrow[48 + m];
        c0 = __builtin_amdgcn_wmma_f32_16x16x4_f32(false, a, false, b0, (short)0, c0, false, false);
        c1 = __builtin_amdgcn_wmma_f32_16x16x4_f32(false, a, false, b1, (short)0, c1, false, false);
        c2 = __builtin_amdgcn_wmma_f32_16x16x4_f32(false, a, false, b2, (short)0, c2, false, false);
        c3 = __builtin_amdgcn_wmma_f32_16x16x4_f32(false, a, false, b3, (short)0, c3, false, false);
    }

    // bias add + spill D tiles to per-wave LDS tile (C/D layout: VGPR r = row mB+r, lane&15 = col)
    int mB = (lane < 16) ? 0 : 8;
    float bb0 = bias[m], bb1 = bias[16 + m], bb2 = bias[32 + m], bb3 = bias[48 + m];
    #pragma unroll
    for (int r = 0; r < 8; ++r) {
        tile[wave][mB + r][m]      = c0[r] + bb0;
        tile[wave][mB + r][16 + m] = c1[r] + bb1;
        tile[wave][mB + r][32 + m] = c2[r] + bb2;
        tile[wave][mB + r][48 + m] = c3[r] + bb3;
    }

    // fused LayerNorm: lanes 0-15 each own one row of the 16-row tile
    // (wave-local DS ops are in-order; compiler inserts s_wait_dscnt)
    if (lane < 16) {
        long row = rowBase + lane;
        if (row < N) {
            float s = 0.f, s2 = 0.f;
            #pragma unroll
            for (int f = 0; f < DFEAT; ++f) {
                float v = tile[wave][lane][f];
                s += v; s2 += v * v;
            }
            float mu  = s  * (1.0f / DFEAT);
            float var = s2 * (1.0f / DFEAT) - mu * mu;
            float inv = rsqrtf(var + 1e-5f);
            float* o = out + row * DFEAT;
            #pragma unroll
            for (int f = 0; f < DFEAT; ++f)
                o[f] = (tile[wave][lane][f] - mu) * inv * g[f] + be[f];
        }
    }
}

// ---------------- launcher ----------------
extern "C" void kernel_launch(void* const* d_in, const int* in_sizes, int n_in,
                              void* d_out, int out_size, void* d_ws, size_t ws_size,
                              hipStream_t stream) {
    const float* node_attr = (const float*)d_in[0];
    const int*   edge_idx  = (const int*)d_in[1];
    // d_in[2] = batch_idx (unused)
    const float* adv0 = (const float*)d_in[3];
    const float* adv1 = (const float*)d_in[4];
    const float* W0 = (const float*)d_in[5];
    const float* b0 = (const float*)d_in[6];
    const float* g0 = (const float*)d_in[7];
    const float* be0 = (const float*)d_in[8];
    const float* W1 = (const float*)d_in[9];
    const float* b1 = (const float*)d_in[10];
    const float* g1 = (const float*)d_in[11];
    const float* be1 = (const float*)d_in[12];

    const int N = in_sizes[0] / DFEAT;
    const int E = in_sizes[3];
    const int* src = edge_idx;        // edge_index[0]
    const int* dst = edge_idx + E;    // edge_index[1]

    float* ws    = (float*)d_ws;
    float* denom = ws;                       ws += N;
    float* aggr  = ws;                       ws += (size_t)N * DFEAT;
    float* out1  = ws;

    float* outF = (float*)d_out;
    const int nd = N * DFEAT;
    const int tiles = (N + 15) / 16;

    auto run_layer = [&](const float* xin, const float* adv, const float* W,
                         const float* b, const float* g, const float* be, float* xout) {
        int n4 = N / 4, nd4 = nd / 4;
        zero_f32x4<<<(n4 + 255) / 256, 256, 0, stream>>>((float4*)denom, n4);
        zero_f32x4<<<(nd4 + 255) / 256, 256, 0, stream>>>((float4*)aggr, nd4);
        edge_denom<<<(E + 255) / 256, 256, 0, stream>>>(dst, adv, denom, E);
        long t16 = (long)E * 16;
        edge_scatter<<<(unsigned)((t16 + 255) / 256), 256, 0, stream>>>(src, dst, adv, denom, xin, aggr, E);
        gemm_bias_ln<<<(tiles + 7) / 8, 256, 0, stream>>>(aggr, xin, W, b, g, be, xout, N);
    };

    run_layer(node_attr, adv0, W0, b0, g0, be0, out1);
    run_layer(out1,      adv1, W1, b1, g1, be1, outF);
}